// STAGATE_31585189495025
// MI455X (gfx1250) — compile-verified
//
#include <hip/hip_runtime.h>

// ---------------- problem constants (from reference) ----------------
#define N_NODES 50000
#define IN_DIM  3000
#define HID     512
#define LAT     30
#define N_EDGES 800000
#define NEG_SLOPE 0.2f

// order-preserving float->u32 key of -INFINITY: bits(-inf)=0xFF800000 (negative) -> ~bits = 0x007FFFFF
#define NEG_INF_FKEY 0x007FFFFFu

typedef __attribute__((ext_vector_type(2))) float v2f;
typedef __attribute__((ext_vector_type(8))) float v8f;

// ---------------- small utility kernels ----------------
__global__ void fill_u32_kernel(unsigned* __restrict__ p, unsigned v, long long n) {
    long long i = (long long)blockIdx.x * blockDim.x + threadIdx.x;
    if (i < n) p[i] = v;
}
__global__ void zero_f32_kernel(float* __restrict__ p, long long n) {
    long long i = (long long)blockIdx.x * blockDim.x + threadIdx.x;
    if (i < n) p[i] = 0.0f;
}
__global__ void elu_inplace_kernel(float* __restrict__ p, long long n) {
    long long i = (long long)blockIdx.x * blockDim.x + threadIdx.x;
    if (i < n) {
        float v = p[i];
        p[i] = (v > 0.0f) ? v : (__expf(v) - 1.0f);
    }
}

// ---------------- fp32 WMMA GEMM: C[M,N] = A[M,K] * B ----------------
// TRANSB=false: B is [K,N] row-major (ldb = row stride of B)
// TRANSB=true : B is [N,K] row-major (ldb = row stride of B), i.e. C = A * B^T
// Each wave computes a 16(M) x 64(N) slab with 4 f32 accumulators,
// stepping K by 4 via V_WMMA_F32_16X16X4_F32.
template <bool TRANSB, bool NGUARD, bool KGUARD>
__global__ void gemm_f32_wmma(const float* __restrict__ A,
                              const float* __restrict__ B,
                              float* __restrict__ C,
                              int M, int N, int K, int lda, int ldb, int ldc) {
    const int wavesPerBlock = blockDim.x >> 5;
    const int wid  = blockIdx.x * wavesPerBlock + (threadIdx.x >> 5);
    const int lane = threadIdx.x & 31;
    const int nQuads = (N + 63) >> 6;
    const int mTile  = wid / nQuads;
    const int nQuad  = wid - mTile * nQuads;
    if (mTile * 16 >= M) return;  // uniform per wave -> EXEC all-ones inside

    const int hi    = lane >> 4;          // 0: lanes 0-15, 1: lanes 16-31
    const int khalf = hi << 1;            // K sub-offset 0 or 2
    const int row   = mTile * 16 + (lane & 15);
    const int col0  = nQuad * 64 + (lane & 15);

    v8f acc0 = {}, acc1 = {}, acc2 = {}, acc3 = {};
    const float* __restrict__ Arow = A + (size_t)row * (size_t)lda;

    for (int k = 0; k < K; k += 4) {
        // ---- A fragment: lane holds A[row][k+khalf], A[row][k+khalf+1]
        v2f a;
        if (KGUARD) {
            a.x = (k + khalf + 0 < K) ? Arow[k + khalf + 0] : 0.0f;
            a.y = (k + khalf + 1 < K) ? Arow[k + khalf + 1] : 0.0f;
        } else {
            a = *(const v2f*)(Arow + k + khalf);
            __builtin_prefetch(Arow + k + 64, 0, 0);   // global_prefetch_b8 on streamed A
        }
        // ---- B fragments for 4 adjacent 16-wide column tiles
        v2f b0, b1, b2, b3;
#define LOAD_B(bt, t)                                                          \
        {                                                                      \
            int col = col0 + (t) * 16;                                         \
            bool cok = (!NGUARD) || (col < N);                                 \
            if (TRANSB) {                                                      \
                const float* Bp = B + (size_t)col * (size_t)ldb + k + khalf;   \
                if (KGUARD) {                                                  \
                    bt.x = (cok && (k + khalf + 0 < K)) ? Bp[0] : 0.0f;        \
                    bt.y = (cok && (k + khalf + 1 < K)) ? Bp[1] : 0.0f;        \
                } else if (cok) {                                              \
                    bt = *(const v2f*)Bp;                                      \
                } else { bt.x = 0.0f; bt.y = 0.0f; }                           \
            } else {                                                           \
                if (KGUARD) {                                                  \
                    bt.x = (cok && (k + khalf + 0 < K))                        \
                         ? B[(size_t)(k + khalf + 0) * ldb + col] : 0.0f;      \
                    bt.y = (cok && (k + khalf + 1 < K))                        \
                         ? B[(size_t)(k + khalf + 1) * ldb + col] : 0.0f;      \
                } else {                                                       \
                    bt.x = cok ? B[(size_t)(k + khalf + 0) * ldb + col] : 0.0f;\
                    bt.y = cok ? B[(size_t)(k + khalf + 1) * ldb + col] : 0.0f;\
                }                                                              \
            }                                                                  \
        }
        LOAD_B(b0, 0) LOAD_B(b1, 1) LOAD_B(b2, 2) LOAD_B(b3, 3)
#undef LOAD_B
        acc0 = __builtin_amdgcn_wmma_f32_16x16x4_f32(false, a, false, b0, (short)0, acc0, false, false);
        acc1 = __builtin_amdgcn_wmma_f32_16x16x4_f32(false, a, false, b1, (short)0, acc1, false, false);
        acc2 = __builtin_amdgcn_wmma_f32_16x16x4_f32(false, a, false, b2, (short)0, acc2, false, false);
        acc3 = __builtin_amdgcn_wmma_f32_16x16x4_f32(false, a, false, b3, (short)0, acc3, false, false);
    }

    // ---- store: C/D layout: VGPR r -> M = mTile*16 + r + 8*hi, N = col tile + lane%16
    const int rbase = mTile * 16 + (hi << 3);
#pragma unroll
    for (int r = 0; r < 8; ++r) {
        size_t rowOff = (size_t)(rbase + r) * (size_t)ldc;
        int c0 = col0;
        if (!NGUARD || c0 < N)      C[rowOff + c0]      = acc0[r];
        if (!NGUARD || c0 + 16 < N) C[rowOff + c0 + 16] = acc1[r];
        if (!NGUARD || c0 + 32 < N) C[rowOff + c0 + 32] = acc2[r];
        if (!NGUARD || c0 + 48 < N) C[rowOff + c0 + 48] = acc3[r];
    }
}

// ---------------- attention vector dots: a_src/a_dst = h1 . att ----------------
__global__ void attn_dots_kernel(const float* __restrict__ h1,
                                 const float* __restrict__ att_src,
                                 const float* __restrict__ att_dst,
                                 float* __restrict__ a_src,
                                 float* __restrict__ a_dst) {
    int wid  = (blockIdx.x * blockDim.x + threadIdx.x) >> 5;
    int lane = threadIdx.x & 31;
    if (wid >= N_NODES) return;
    const float* row = h1 + (size_t)wid * HID;
    float s0 = 0.0f, s1 = 0.0f;
    for (int c = lane; c < HID; c += 32) {
        float v = row[c];
        s0 += v * att_src[c];
        s1 += v * att_dst[c];
    }
#pragma unroll
    for (int off = 16; off > 0; off >>= 1) {
        s0 += __shfl_down(s0, off, 32);
        s1 += __shfl_down(s1, off, 32);
    }
    if (lane == 0) { a_src[wid] = s0; a_dst[wid] = s1; }
}

// ---------------- edge softmax: max / exp-sum ----------------
__device__ __forceinline__ unsigned fkey(float f) {
    unsigned u = __float_as_uint(f);
    return (u & 0x80000000u) ? ~u : (u | 0x80000000u);   // order-preserving map
}
__device__ __forceinline__ float funkey(unsigned k) {
    unsigned u = (k & 0x80000000u) ? (k & 0x7FFFFFFFu) : ~k;
    return __uint_as_float(u);
}

__global__ void edge_score_max_kernel(const int* __restrict__ src,
                                      const int* __restrict__ dst,
                                      const float* __restrict__ a_src,
                                      const float* __restrict__ a_dst,
                                      float* __restrict__ score,
                                      unsigned* __restrict__ mkey) {
    long long i = (long long)blockIdx.x * blockDim.x + threadIdx.x;
    if (i >= N_EDGES) return;
    float e = a_src[src[i]] + a_dst[dst[i]];
    e = (e > 0.0f) ? e : NEG_SLOPE * e;          // leaky_relu
    score[i] = e;
    atomicMax(&mkey[dst[i]], fkey(e));
}

__global__ void edge_exp_sum_kernel(const int* __restrict__ dst,
                                    const unsigned* __restrict__ mkey,
                                    float* __restrict__ score,   // in: e, out: exp(e-m)
                                    float* __restrict__ ssum) {
    long long i = (long long)blockIdx.x * blockDim.x + threadIdx.x;
    if (i >= N_EDGES) return;
    int d = dst[i];
    float m = funkey(mkey[d]);
    if (!(m > -3.0e38f)) m = 0.0f;               // empty-segment guard (ref: isfinite)
    float p = expf(score[i] - m);
    score[i] = p;
    atomicAdd(&ssum[d], p);
}

// ---------------- weighted scatter-add: agg[dst] += h[src] * alpha ----------------
__global__ void scatter_agg_kernel(const int* __restrict__ src,
                                   const int* __restrict__ dst,
                                   const float* __restrict__ p,
                                   const float* __restrict__ ssum,
                                   const float* __restrict__ h,
                                   float* __restrict__ agg,
                                   long long total) {
    long long i = (long long)blockIdx.x * blockDim.x + threadIdx.x;
    if (i >= total) return;
    long long e = i >> 9;                 // / HID (512)
    int c = (int)(i & (HID - 1));
    int d = dst[e];
    float alpha = p[e] / (ssum[d] + 1e-16f);
    float v = h[(size_t)src[e] * HID + c] * alpha;
    atomicAdd(&agg[(size_t)d * HID + c], v);     // L2-resident (agg = 102 MB < 192 MB)
}

// ---------------- host-side orchestration ----------------
static inline unsigned cdiv_u(long long a, long long b) { return (unsigned)((a + b - 1) / b); }

extern "C" void kernel_launch(void* const* d_in, const int* in_sizes, int n_in,
                              void* d_out, int out_size, void* d_ws, size_t ws_size,
                              hipStream_t stream) {
    (void)in_sizes; (void)n_in; (void)out_size; (void)ws_size;

    const float* x       = (const float*)d_in[0];
    const int*   eidx    = (const int*)d_in[1];     // [2, E]; harness convention: int32
    const float* W1      = (const float*)d_in[2];   // [IN_DIM, HID]
    const float* W2      = (const float*)d_in[3];   // [HID, LAT]
    const float* att_src = (const float*)d_in[4];   // [HID]
    const float* att_dst = (const float*)d_in[5];   // [HID]
    const int* src = eidx;
    const int* dst = eidx + N_EDGES;

    float* out_x2 = (float*)d_out;                          // [N_NODES, LAT]
    float* out_x4 = (float*)d_out + (size_t)N_NODES * LAT;  // [N_NODES, IN_DIM]

    // workspace carve-up (floats)
    const size_t NH = (size_t)N_NODES * HID;                // 25.6M
    float* bufH  = (float*)d_ws;                // h1, later h3
    float* bufX  = bufH + NH;                   // agg1/x1, later agg3/x3
    float* aSrc  = bufX + NH;
    float* aDst  = aSrc + N_NODES;
    unsigned* mkey = (unsigned*)(aDst + N_NODES);
    float* ssum  = (float*)(mkey + N_NODES);
    float* score = ssum + N_NODES;              // per-edge e -> p

    const dim3 T(256);
    const int WPB = 256 / 32;                   // waves per block

    // init
    fill_u32_kernel<<<cdiv_u(N_NODES, 256), T, 0, stream>>>(mkey, NEG_INF_FKEY, N_NODES);
    zero_f32_kernel<<<cdiv_u(N_NODES, 256), T, 0, stream>>>(ssum, N_NODES);
    zero_f32_kernel<<<cdiv_u((long long)NH, 256), T, 0, stream>>>(bufX, (long long)NH);

    // ---- conv1 GEMM: h1 = x @ W1   [50000,3000]x[3000,512]
    {
        unsigned waves = (N_NODES / 16) * (HID / 64);   // 3125 * 8
        gemm_f32_wmma<false, false, false><<<cdiv_u(waves, WPB), T, 0, stream>>>(
            x, W1, bufH, N_NODES, HID, IN_DIM, IN_DIM, HID, HID);
    }

    // ---- attention scalars + edge softmax (tied alpha)
    attn_dots_kernel<<<cdiv_u((long long)N_NODES, WPB), T, 0, stream>>>(bufH, att_src, att_dst, aSrc, aDst);
    edge_score_max_kernel<<<cdiv_u(N_EDGES, 256), T, 0, stream>>>(src, dst, aSrc, aDst, score, mkey);
    edge_exp_sum_kernel  <<<cdiv_u(N_EDGES, 256), T, 0, stream>>>(dst, mkey, score, ssum);

    // ---- agg1 = segsum(h1[src]*alpha, dst); x1 = elu(agg1)
    {
        long long total = (long long)N_EDGES * HID;     // 409.6M
        scatter_agg_kernel<<<cdiv_u(total, 256), T, 0, stream>>>(src, dst, score, ssum, bufH, bufX, total);
        elu_inplace_kernel<<<cdiv_u((long long)NH, 256), T, 0, stream>>>(bufX, (long long)NH);
    }

    // ---- conv2: x2 = x1 @ W2   [50000,512]x[512,30]  (N-guarded)
    {
        unsigned waves = (N_NODES / 16) * 1;            // nQuads = ceil(30/64) = 1
        gemm_f32_wmma<false, true, false><<<cdiv_u(waves, WPB), T, 0, stream>>>(
            bufX, W2, out_x2, N_NODES, LAT, HID, HID, LAT, LAT);
    }

    // ---- conv3: h3 = x2 @ W2^T  [50000,30]x[30,512]  (K-guarded, B transposed)
    {
        unsigned waves = (N_NODES / 16) * (HID / 64);
        gemm_f32_wmma<true, false, true><<<cdiv_u(waves, WPB), T, 0, stream>>>(
            out_x2, W2, bufH, N_NODES, HID, LAT, LAT, LAT, HID);
    }

    // ---- agg3 (same tied alpha); x3 = elu(agg3)
    zero_f32_kernel<<<cdiv_u((long long)NH, 256), T, 0, stream>>>(bufX, (long long)NH);
    {
        long long total = (long long)N_EDGES * HID;
        scatter_agg_kernel<<<cdiv_u(total, 256), T, 0, stream>>>(src, dst, score, ssum, bufH, bufX, total);
        elu_inplace_kernel<<<cdiv_u((long long)NH, 256), T, 0, stream>>>(bufX, (long long)NH);
    }

    // ---- conv4: x4 = x3 @ W1^T  [50000,512]x[512,3000]  (B transposed, N-guarded)
    {
        unsigned nQuads = (IN_DIM + 63) / 64;           // 47
        unsigned waves = (N_NODES / 16) * nQuads;
        gemm_f32_wmma<true, true, false><<<cdiv_u(waves, WPB), T, 0, stream>>>(
            bufX, W1, out_x4, N_NODES, IN_DIM, HID, HID, HID, IN_DIM);
    }
}